// LQRPrediction_61108794687783
// MI455X (gfx1250) — compile-verified
//
#include <hip/hip_runtime.h>
#include <cstdint>

typedef __attribute__((ext_vector_type(2))) float v2f;
typedef __attribute__((ext_vector_type(8))) float v8f;

#define HH 100
#define BB 256
#define NS 32
#define NC 16
#define NT 48

// padded LDS row strides (floats): spread banks, keep 8/16-byte alignment
#define FTS 36  // sFT: F^T   (48 x 32)  double-buffered
#define KTS 20  // sKT: K^T   (32 x 16)  double-buffered
#define VTS 36  // sVT: V^T   (32 x 32)  carry
#define FVS 34  // sFV: F^T V (48 x 32) row-major
#define QS  52  // sQ : Q     (48 x 48) row-major
#define QTS 52  // sQT: Q^T   (48 x 48)
#define PTS 20  // sPT: P^T   (32 x 16), P = Qux + Quu K (16 x 32)

static __device__ __forceinline__ v8f wmma4(v2f a, v2f b, v8f c) {
  // v_wmma_f32_16x16x4_f32 : D = A(16x4) * B(4x16) + C
  return __builtin_amdgcn_wmma_f32_16x16x4_f32(false, a, false, b, (short)0, c, false, false);
}
static __device__ __forceinline__ v2f ld2(const float* p) { return *(const v2f*)p; }
static __device__ __forceinline__ v8f seed8(const float* p) {
  float4 s0 = *(const float4*)p;
  float4 s1 = *(const float4*)(p + 4);
  v8f a;
  a[0] = s0.x; a[1] = s0.y; a[2] = s0.z; a[3] = s0.w;
  a[4] = s1.x; a[5] = s1.y; a[6] = s1.z; a[7] = s1.w;
  return a;
}
static __device__ __forceinline__ void st8(float* p, v8f a) {
  float4 s0, s1;
  s0.x = a[0]; s0.y = a[1]; s0.z = a[2]; s0.w = a[3];
  s1.x = a[4]; s1.y = a[5]; s1.z = a[6]; s1.w = a[7];
  *(float4*)p = s0; *(float4*)(p + 4) = s1;
}

// ---- CDNA5 async LDS<->global DMA (ASYNCcnt-tracked) ----
static __device__ __forceinline__ void async_ld_b32(float* ldsp, uint32_t goff,
                                                    const float* gbase) {
  asm volatile("global_load_async_to_lds_b32 %0, %1, %2"
               :
               : "v"((uint32_t)(uintptr_t)ldsp), "v"(goff),
                 "s"((uint64_t)(uintptr_t)gbase)
               : "memory");
}
static __device__ __forceinline__ void async_st_b128(uint32_t goff, const float* ldsp,
                                                     float* gbase) {
  asm volatile("global_store_async_from_lds_b128 %0, %1, %2"
               :
               : "v"(goff), "v"((uint32_t)(uintptr_t)ldsp),
                 "s"((uint64_t)(uintptr_t)gbase)
               : "memory");
}
static __device__ __forceinline__ void wait_async0() {
  asm volatile("s_wait_asynccnt 0x0" ::: "memory");
}

// stage F (32x48 -> F^T), K (16x32 -> K^T), and f, c, k vectors via async DMA
static __device__ __forceinline__ void stage_async(
    int tid, const float* gFs, const float* gfs, const float* gcs,
    const float* gKs, const float* gks,
    float* dFT, float* dKT, float* df, float* dc, float* dk) {
  for (int j = 0; j < 6; ++j) {  // 1536 elems of F, transposed scatter
    const int i = tid + 256 * j;
    const int r = i / 48, c = i % 48;
    async_ld_b32(&dFT[c * FTS + r], 4u * (uint32_t)i, gFs);
  }
  for (int j = 0; j < 2; ++j) {  // 512 elems of K, transposed scatter
    const int i = tid + 256 * j;
    const int r = i / 32, c = i % 32;
    async_ld_b32(&dKT[c * KTS + r], 4u * (uint32_t)i, gKs);
  }
  if (tid < NS) async_ld_b32(&df[tid], 4u * (uint32_t)tid, gfs);
  if (tid < NT) async_ld_b32(&dc[tid], 4u * (uint32_t)tid, gcs);
  if (tid < NC) async_ld_b32(&dk[tid], 4u * (uint32_t)tid, gks);
}

__global__ __launch_bounds__(256) void lqr_pred_kernel(
    const float* __restrict__ gF, const float* __restrict__ gf,
    const float* __restrict__ gC, const float* __restrict__ gc,
    const float* __restrict__ gK, const float* __restrict__ gk,
    float* __restrict__ out) {
  __shared__ __align__(16) float sFT[2][NT * FTS];
  __shared__ __align__(16) float sKT[2][NS * KTS];
  __shared__ __align__(16) float sVT[NS * VTS];
  __shared__ __align__(16) float sFV[NT * FVS];
  __shared__ __align__(16) float sQ[NT * QS];
  __shared__ __align__(16) float sQT[NT * QTS];
  __shared__ __align__(16) float sPT[NS * PTS];
  __shared__ float sfB[2][NS], scB[2][NT], skB[2][NC];
  __shared__ float sv[NS], sq[NT], sw1[NS], sT2[NC];
  __shared__ float sScal[2];  // [0]=qc (this step), [1]=vc (carry)

  const int tid  = threadIdx.x;
  const int b    = blockIdx.x;
  const int lane = tid & 31;
  const int wid  = tid >> 5;
  const int lm   = lane & 15;
  const int lg   = lane >> 4;

  // output section offsets (floats); tuple order: Qs,qs,qcs,Vs,vs,vcs
  const size_t OFF_q  = (size_t)HH * BB * NT * NT;
  const size_t OFF_qc = OFF_q + (size_t)HH * BB * NT;
  const size_t OFF_V  = OFF_qc + (size_t)HH * BB;
  const size_t OFF_v  = OFF_V + (size_t)(HH + 1) * BB * NS * NS;
  const size_t OFF_vc = OFF_v + (size_t)(HH + 1) * BB * NS;

  // init carries: V^T = 0, v = 0, vc = 0
  for (int i = tid; i < NS * VTS; i += 256) sVT[i] = 0.f;
  if (tid < NS) sv[tid] = 0.f;
  if (tid == 0) sScal[1] = 0.f;

  // preload step t = HH-1 into buffer 0 (async DMA to LDS)
  {
    const size_t sl0 = (size_t)(HH - 1) * BB + b;
    stage_async(tid, gF + sl0 * (NS * NT), gf + sl0 * NS, gc + sl0 * NT,
                gK + sl0 * (NC * NS), gk + sl0 * NC,
                sFT[0], sKT[0], sfB[0], scB[0], skB[0]);
    __builtin_prefetch(gC + sl0 * (NT * NT) + tid * 9, 0, 0);
    wait_async0();
  }
  __syncthreads();

  for (int t = HH - 1; t >= 0; --t) {
    const size_t sl = (size_t)t * BB + b;
    const int cur = (HH - 1 - t) & 1;
    const float* FT = sFT[cur];
    const float* KT = sKT[cur];
    const float* pf = sfB[cur];
    const float* pc = scB[cur];
    const float* pk = skB[cur];

    // ---- issue async staging of step t-1 into the other buffer ----
    if (t > 0) {
      const size_t sn = sl - BB;
      const int nx = cur ^ 1;
      stage_async(tid, gF + sn * (NS * NT), gf + sn * NS, gc + sn * NT,
                  gK + sn * (NC * NS), gk + sn * NC,
                  sFT[nx], sKT[nx], sfB[nx], scB[nx], skB[nx]);
      __builtin_prefetch(gC + sn * (NT * NT) + tid * 9, 0, 0);  // C read direct
    }

    // ---- phase A: FV = F^T @ V (48x32, 6 tiles, waves 0..5); wave6: w1 = V f
    if (wid < 6) {
      const int tr = wid >> 1, tc = wid & 1;
      v8f acc = {};
      for (int k0 = 0; k0 < NS; k0 += 4) {
        v2f a = ld2(&FT[(tr * 16 + lm) * FTS + k0 + 2 * lg]);   // (F^T) rows
        v2f bm = ld2(&sVT[(tc * 16 + lm) * VTS + k0 + 2 * lg]); // V cols
        acc = wmma4(a, bm, acc);
      }
      for (int r = 0; r < 8; ++r)
        sFV[(tr * 16 + r + 8 * lg) * FVS + tc * 16 + lm] = acc[r];
    } else if (wid == 6) {
      float s = 0.f;
      for (int j = 0; j < NS; ++j) s += sVT[j * VTS + lane] * pf[j];
      sw1[lane] = s;  // w1 = V f
    }
    __syncthreads();

    // ---- phase B: Q = C + FV @ F (9 tiles over 8 waves); then q, qc ----
    const float* gCt = gC + sl * (NT * NT);
    for (int tile = wid; tile < 9; tile += 8) {
      const int tr = tile / 3, tc = tile % 3;
      float cs[8];
      for (int r = 0; r < 8; ++r)  // C tile direct from global (no reuse)
        cs[r] = gCt[(tr * 16 + r + 8 * lg) * NT + tc * 16 + lm];
      v8f acc = {};
      for (int k0 = 0; k0 < NS; k0 += 4) {
        v2f a = ld2(&sFV[(tr * 16 + lm) * FVS + k0 + 2 * lg]);  // FV rows
        v2f bm = ld2(&FT[(tc * 16 + lm) * FTS + k0 + 2 * lg]);  // F cols
        acc = wmma4(a, bm, acc);
      }
      for (int r = 0; r < 8; ++r) acc[r] += cs[r];
      st8(&sQT[(tc * 16 + lm) * QTS + tr * 16 + 8 * lg], acc);  // Q^T copy
      for (int r = 0; r < 8; ++r)
        sQ[(tr * 16 + r + 8 * lg) * QS + tc * 16 + lm] = acc[r];
    }
    if (tid < NT) {  // q = c + FV f + F^T v
      float s = pc[tid];
      for (int j = 0; j < NS; ++j) s += sFV[tid * FVS + j] * pf[j];
      for (int j = 0; j < NS; ++j) s += FT[tid * FTS + j] * sv[j];
      sq[tid] = s;
      out[OFF_q + sl * NT + tid] = s;
    }
    if (tid == 0) {  // qc = f.w1/2 + f.v + vc
      float s = 0.f, s2 = 0.f;
      for (int i = 0; i < NS; ++i) { s += pf[i] * sw1[i]; s2 += pf[i] * sv[i]; }
      const float qc = 0.5f * s + s2 + sScal[1];
      sScal[0] = qc;
      out[OFF_qc + sl] = qc;
    }
    __syncthreads();

    // ---- phase C: P = Qux + Quu @ K (waves 0..1); wave2: T2 = Quu k;
    //      all threads: async DMA Q -> global from LDS
    if (wid < 2) {
      const int tc = wid;
      v8f acc = seed8(&sQT[(tc * 16 + lm) * QTS + NS + 8 * lg]);  // Qux tile
      for (int k0 = 0; k0 < NC; k0 += 4) {
        v2f a = ld2(&sQ[(NS + lm) * QS + NS + k0 + 2 * lg]);      // Quu rows
        v2f bm = ld2(&KT[(tc * 16 + lm) * KTS + k0 + 2 * lg]);    // K cols
        acc = wmma4(a, bm, acc);
      }
      st8(&sPT[(tc * 16 + lm) * PTS + 8 * lg], acc);              // P^T
    } else if (wid == 2) {
      if (lane < NC) {
        float s = 0.f;
        for (int j = 0; j < NC; ++j) s += sQ[(NS + lane) * QS + NS + j] * pk[j];
        sT2[lane] = s;  // T2 = Quu k
      }
    }
    {
      float* gQ = out + sl * (NT * NT);
      for (int i = tid; i < NT * NT / 4; i += 256) {  // LDS -> global DMA
        const int r = i / 12, c = (i % 12) * 4;
        async_st_b128(4u * (uint32_t)(r * NT + c), &sQ[r * QS + c], gQ);
      }
    }
    __syncthreads();

    // ---- phase D: Vn = Qxx + Qxu @ K + K^T @ P (4 tiles, waves 0..3) ----
    float* gV = out + OFF_V + sl * (NS * NS);
    if (wid < 4) {
      const int tr = wid >> 1, tc = wid & 1;
      v8f acc = seed8(&sQT[(tc * 16 + lm) * QTS + tr * 16 + 8 * lg]);  // Qxx
      for (int k0 = 0; k0 < NC; k0 += 4) {  // += Qxu @ K
        v2f a = ld2(&sQ[(tr * 16 + lm) * QS + NS + k0 + 2 * lg]);
        v2f bm = ld2(&KT[(tc * 16 + lm) * KTS + k0 + 2 * lg]);
        acc = wmma4(a, bm, acc);
      }
      for (int k0 = 0; k0 < NC; k0 += 4) {  // += K^T @ P
        v2f a = ld2(&KT[(tr * 16 + lm) * KTS + k0 + 2 * lg]);
        v2f bm = ld2(&sPT[(tc * 16 + lm) * PTS + k0 + 2 * lg]);
        acc = wmma4(a, bm, acc);
      }
      st8(&sVT[(tc * 16 + lm) * VTS + tr * 16 + 8 * lg], acc);  // next carry V^T
      for (int r = 0; r < 8; ++r)
        gV[(tr * 16 + r + 8 * lg) * NS + tc * 16 + lm] = acc[r];
    } else if (wid == 4) {  // vn = qx + Qxu k + K^T (qu + T2)
      float s = sq[lane];
      for (int j = 0; j < NC; ++j) s += sQ[lane * QS + NS + j] * pk[j];
      for (int j = 0; j < NC; ++j) s += KT[lane * KTS + j] * (sq[NS + j] + sT2[j]);
      sv[lane] = s;  // next carry
      out[OFF_v + sl * NS + lane] = s;
    } else if (wid == 5) {
      if (lane == 0) {  // vcn = k.(T2/2 + qu) + qc
        float s = 0.f;
        for (int i = 0; i < NC; ++i) s += pk[i] * (0.5f * sT2[i] + sq[NS + i]);
        const float vcn = s + sScal[0];
        sScal[1] = vcn;  // next carry
        out[OFF_vc + sl] = vcn;
      }
    }
    // drain this step's async staging + Q writeback, then release buffers
    wait_async0();
    __syncthreads();
  }

  // trailing zero blocks: Vs[H], vs[H], vcs[H]
  {
    const size_t slH = (size_t)HH * BB + b;
    float* gVH = out + OFF_V + slH * (NS * NS);
    for (int i = tid; i < NS * NS; i += 256) gVH[i] = 0.f;
    if (tid < NS) out[OFF_v + slH * NS + tid] = 0.f;
    if (tid == 0) out[OFF_vc + slH] = 0.f;
  }
}

extern "C" void kernel_launch(void* const* d_in, const int* in_sizes, int n_in,
                              void* d_out, int out_size, void* d_ws, size_t ws_size,
                              hipStream_t stream) {
  (void)in_sizes; (void)n_in; (void)out_size; (void)d_ws; (void)ws_size;
  const float* F = (const float*)d_in[0];
  const float* f = (const float*)d_in[1];
  const float* C = (const float*)d_in[2];
  const float* c = (const float*)d_in[3];
  const float* K = (const float*)d_in[4];
  const float* k = (const float*)d_in[5];
  float* out = (float*)d_out;
  lqr_pred_kernel<<<BB, 256, 0, stream>>>(F, f, C, c, K, k, out);
}